// LPF_84894323572807
// MI455X (gfx1250) — compile-verified
//
#include <hip/hip_runtime.h>

typedef float v2f __attribute__((ext_vector_type(2)));
typedef float v8f __attribute__((ext_vector_type(8)));

#define B_    64
#define C_    32
#define HW_   1024
#define NPIX  (B_ * HW_)     /* 65536 */
#define NTOT  (NPIX * C_)    /* 2097152 */
#define SLOPE 0.05f

// ---------------------------------------------------------------------------
// Conv 3x3 SAME (32->32), NCHW input -> (B, N, C) "pixel-major" output
// one block per (b, h) row; weights + 3 input rows staged in LDS
// ---------------------------------------------------------------------------
__global__ __launch_bounds__(256) void conv_kernel(const float* __restrict__ x,
                                                   const float* __restrict__ w,
                                                   const float* __restrict__ bias,
                                                   float* __restrict__ z) {
  __shared__ float xs[32 * 3 * 34];   // [ic][kh][col], col = w+1 (zero halo)
  __shared__ float wsl[32 * 32 * 9];  // [oc][ic][kh][kw]
  const int b = blockIdx.x >> 5, h = blockIdx.x & 31, tid = threadIdx.x;
  for (int i = tid; i < 32 * 32 * 9; i += 256) wsl[i] = w[i];
  for (int i = tid; i < 32 * 3 * 34; i += 256) {
    int ic = i / 102, rem = i % 102, kh = rem / 34, col = rem % 34;
    int hh = h + kh - 1, w0 = col - 1;
    float v = 0.f;
    if ((unsigned)hh < 32u && (unsigned)w0 < 32u)
      v = x[(((b * 32 + ic) * 32 + hh) * 32) + w0];
    xs[i] = v;
  }
  __syncthreads();
  for (int o = tid; o < 1024; o += 256) {
    int oc = o & 31, ww = o >> 5;
    float acc = bias[oc];
    const float* wp = &wsl[oc * 288];
    for (int ic = 0; ic < 32; ++ic) {
      const float* xp = &xs[ic * 102];
#pragma unroll
      for (int kh = 0; kh < 3; ++kh)
#pragma unroll
        for (int kw = 0; kw < 3; ++kw)
          acc += xp[kh * 34 + ww + kw] * wp[ic * 9 + kh * 3 + kw];
    }
    z[((size_t)(b * HW_ + h * 32 + ww)) * 32 + oc] = acc;
  }
}

// ---------------------------------------------------------------------------
// Per-channel BN stats over 65536 samples -> affine (A = g*rsqrt(v+eps),
// Bc = b - m*A). One block per channel.
// ---------------------------------------------------------------------------
__global__ __launch_bounds__(256) void stats_kernel(const float* __restrict__ z,
                                                    const float* __restrict__ g,
                                                    const float* __restrict__ bb,
                                                    float* __restrict__ AB) {
  __shared__ double s1[256], s2[256];
  const int c = blockIdx.x, tid = threadIdx.x;
  double a = 0.0, q = 0.0;
  for (int i = tid; i < NPIX; i += 256) {
    float v = z[(size_t)i * 32 + c];
    a += v;
    q += (double)v * v;
  }
  s1[tid] = a; s2[tid] = q;
  __syncthreads();
  for (int s = 128; s > 0; s >>= 1) {
    if (tid < s) { s1[tid] += s1[tid + s]; s2[tid] += s2[tid + s]; }
    __syncthreads();
  }
  if (tid == 0) {
    double m = s1[0] / (double)NPIX;
    double var = s2[0] / (double)NPIX - m * m;
    float A = rsqrtf((float)var + 1e-5f) * g[c];
    AB[c] = A;
    AB[32 + c] = bb[c] - (float)m * A;
  }
}

// ---------------------------------------------------------------------------
// BN affine + leaky relu + per-pixel squared norm (for KNN). One wave = one
// pixel's 32 channels; r reduced via shfl_xor.
// ---------------------------------------------------------------------------
__global__ __launch_bounds__(256) void bnleaky_kernel(const float* __restrict__ z,
                                                      const float* __restrict__ AB,
                                                      float* __restrict__ y,
                                                      float* __restrict__ r) {
  const int tid = threadIdx.x;
  const int pix = blockIdx.x * 8 + (tid >> 5);
  const int c = tid & 31;
  float v = z[(size_t)pix * 32 + c] * AB[c] + AB[32 + c];
  float yv = v >= 0.f ? v : SLOPE * v;
  y[(size_t)pix * 32 + c] = yv;
  float s = yv * yv;
#pragma unroll
  for (int o = 16; o > 0; o >>= 1) s += __shfl_xor(s, o);
  if (c == 0) r[pix] = s;
}

// ---------------------------------------------------------------------------
// Branchless top-8 insertion (ascending by (d, index)). Pure cmp/cndmask —
// no control flow, so the compiler keeps ONE copy of the 16-register list
// (no phi-node v_mov bloat inside the hot gram-tile loop).
// ---------------------------------------------------------------------------
__device__ __forceinline__ bool pair_less(float da, int ia, float db, int ib) {
  return (da < db) | ((da == db) & (ia < ib));
}

__device__ __forceinline__ void topk_insert_nb(float best[8], int bi[8],
                                               float d, int m, bool ins) {
  // conditional replace of the worst element
  bool rep = ins & pair_less(d, m, best[7], bi[7]);
  best[7] = rep ? d : best[7];
  bi[7]   = rep ? m : bi[7];
  // bubble the (possibly) new element to its slot — compare-exchange network
#pragma unroll
  for (int j = 7; j > 0; --j) {
    bool sw = pair_less(best[j], bi[j], best[j - 1], bi[j - 1]);
    float dmin = sw ? best[j] : best[j - 1];
    float dmax = sw ? best[j - 1] : best[j];
    int   imin = sw ? bi[j] : bi[j - 1];
    int   imax = sw ? bi[j - 1] : bi[j];
    best[j - 1] = dmin; best[j] = dmax;
    bi[j - 1]   = imin; bi[j]   = imax;
  }
}

// ---------------------------------------------------------------------------
// KNN: one wave handles 16 query rows of one batch. Gram tiles 16x16 via
// V_WMMA_F32_16X16X4_F32 (D[M=cand][N=query]); d = rq + rc - 2*gram.
// Lane L<16 sees candidates m%16 in 0..7 of query L; lane L+16 sees 8..15;
// per-lane sorted top-8, merged across the lane pair at the end.
// Insert is guarded by a wave-uniform __any() early-out: once every lane's
// top-8 is good, late tiles only pay 8 compares.
// ---------------------------------------------------------------------------
__global__ __launch_bounds__(32) void knn_kernel(const float* __restrict__ y,
                                                 const float* __restrict__ r,
                                                 int* __restrict__ idx) {
  __shared__ float qf[16 * 32];
  __shared__ float cf[16 * 32];
  __shared__ float rcs[16];
  const int b = blockIdx.x >> 6;
  const int q0 = (blockIdx.x & 63) * 16;
  const int lane = threadIdx.x;
  const int lr = lane & 15;
  const int koff = (lane < 16) ? 0 : 2;

  for (int i = 0; i < 16; ++i)
    qf[i * 32 + lane] = y[((size_t)(b * HW_ + q0 + i)) * 32 + lane];
  const float rq = r[b * HW_ + q0 + lr];
  const int q = q0 + lr;

  float best[8]; int bi[8];
#pragma unroll
  for (int j = 0; j < 8; ++j) { best[j] = 3.4e38f; bi[j] = 0x7fffffff; }

  for (int ct = 0; ct < 64; ++ct) {
    __syncthreads();
    for (int i = 0; i < 16; ++i)
      cf[i * 32 + lane] = y[((size_t)(b * HW_ + ct * 16 + i)) * 32 + lane];
    if (lane < 16) rcs[lane] = r[b * HW_ + ct * 16 + lane];
    __syncthreads();

    v8f acc = {};
#pragma unroll
    for (int ks = 0; ks < 8; ++ks) {
      int kk = ks * 4 + koff;
      v2f av, bv;
      av.x = cf[lr * 32 + kk]; av.y = cf[lr * 32 + kk + 1];
      bv.x = qf[lr * 32 + kk]; bv.y = qf[lr * 32 + kk + 1];
      acc = __builtin_amdgcn_wmma_f32_16x16x4_f32(false, av, false, bv,
                                                  (short)0, acc, false, false);
    }

    float dv[8]; int mv[8];
    bool need = false;
#pragma unroll
    for (int v = 0; v < 8; ++v) {
      int ml = (lane < 16) ? v : (v + 8);
      int m = ct * 16 + ml;
      float d = rq + rcs[ml] - 2.f * acc[v];
      d = (m == q) ? 1e30f : d;  // exclude self
      dv[v] = d; mv[v] = m;
      need |= pair_less(d, m, best[7], bi[7]);
    }
    if (__any(need)) {
#pragma unroll
      for (int v = 0; v < 8; ++v)
        topk_insert_nb(best, bi, dv[v], mv[v], true);
    }
  }

  // merge with partner lane (same query, other half of candidates)
  float pd[8]; int pi[8];
#pragma unroll
  for (int j = 0; j < 8; ++j) {
    pd[j] = __shfl_xor(best[j], 16);
    pi[j] = __shfl_xor(bi[j], 16);
  }
#pragma unroll
  for (int j = 0; j < 8; ++j) topk_insert_nb(best, bi, pd[j], pi[j], true);

  if (lane < 16) {
#pragma unroll
    for (int j = 0; j < 8; ++j)
      idx[((size_t)(b * HW_ + q0 + lane)) * 8 + j] = bi[j];
  }
}

// ---------------------------------------------------------------------------
// NLA: out = mean_k(nbr)@Wd + f@(Ws-Wd) + (bd+bs). One wave per 16-pixel tile;
// two chained WMMA GEMM passes (K=32, 8 steps each) per 16-col output tile.
// ---------------------------------------------------------------------------
__global__ __launch_bounds__(32) void nla_kernel(const float* __restrict__ y,
                                                 const int* __restrict__ idx,
                                                 const float* __restrict__ Wd,
                                                 const float* __restrict__ Ws,
                                                 const float* __restrict__ bd,
                                                 const float* __restrict__ bs,
                                                 float* __restrict__ z) {
  __shared__ float ft[16 * 32];
  __shared__ float gt[16 * 32];
  __shared__ float wd[32 * 32];
  __shared__ float wc[32 * 32];
  __shared__ float bc[32];
  const int b = blockIdx.x >> 6;
  const int n0 = (blockIdx.x & 63) * 16;
  const int lane = threadIdx.x;
  const int lr = lane & 15;
  const int koff = (lane < 16) ? 0 : 2;

  for (int i = lane; i < 1024; i += 32) {
    float d = Wd[i];
    wd[i] = d;
    wc[i] = Ws[i] - d;
  }
  bc[lane] = bd[lane] + bs[lane];
  for (int i = 0; i < 16; ++i)
    ft[i * 32 + lane] = y[((size_t)(b * HW_ + n0 + i)) * 32 + lane];
  __syncthreads();

  for (int i = 0; i < 16; ++i) {  // gather-mean of 8 neighbors, lane = channel
    const int* ip = &idx[((size_t)(b * HW_ + n0 + i)) * 8];
    float s = 0.f;
#pragma unroll
    for (int k = 0; k < 8; ++k)
      s += y[((size_t)(b * HW_ + ip[k])) * 32 + lane];
    gt[i * 32 + lane] = s * 0.125f;
  }
  __syncthreads();

#pragma unroll
  for (int ctile = 0; ctile < 2; ++ctile) {
    const int col = ctile * 16 + lr;
    v8f acc = {};
#pragma unroll
    for (int ks = 0; ks < 8; ++ks) {  // g @ Wd
      int kk = ks * 4 + koff;
      v2f av, bv;
      av.x = gt[lr * 32 + kk];      av.y = gt[lr * 32 + kk + 1];
      bv.x = wd[kk * 32 + col];     bv.y = wd[(kk + 1) * 32 + col];
      acc = __builtin_amdgcn_wmma_f32_16x16x4_f32(false, av, false, bv,
                                                  (short)0, acc, false, false);
    }
#pragma unroll
    for (int ks = 0; ks < 8; ++ks) {  // f @ (Ws - Wd), chained into acc
      int kk = ks * 4 + koff;
      v2f av, bv;
      av.x = ft[lr * 32 + kk];      av.y = ft[lr * 32 + kk + 1];
      bv.x = wc[kk * 32 + col];     bv.y = wc[(kk + 1) * 32 + col];
      acc = __builtin_amdgcn_wmma_f32_16x16x4_f32(false, av, false, bv,
                                                  (short)0, acc, false, false);
    }
    const float bias = bc[col];
#pragma unroll
    for (int v = 0; v < 8; ++v) {
      int row = (lane < 16) ? v : (v + 8);
      z[((size_t)(b * HW_ + n0 + row)) * 32 + col] = acc[v] + bias;
    }
  }
}

// ---------------------------------------------------------------------------
// Final: BN + leaky + residual, (B,N,C) -> NCHW
// ---------------------------------------------------------------------------
__global__ __launch_bounds__(256) void final_kernel(const float* __restrict__ x,
                                                    const float* __restrict__ z,
                                                    const float* __restrict__ AB,
                                                    float* __restrict__ out) {
  const int t = blockIdx.x * 256 + threadIdx.x;
  const int b = t >> 15;
  const int c = (t >> 10) & 31;
  const int n = t & 1023;
  float v = z[((size_t)(b * HW_ + n)) * 32 + c] * AB[c] + AB[32 + c];
  float yv = v >= 0.f ? v : SLOPE * v;
  out[t] = x[t] + yv;
}

extern "C" void kernel_launch(void* const* d_in, const int* in_sizes, int n_in,
                              void* d_out, int out_size, void* d_ws, size_t ws_size,
                              hipStream_t stream) {
  const float* x      = (const float*)d_in[0];
  const float* conv_w = (const float*)d_in[1];
  const float* conv_b = (const float*)d_in[2];
  const float* g[4]   = {(const float*)d_in[3], (const float*)d_in[9],
                         (const float*)d_in[15], (const float*)d_in[21]};
  const float* bn[4]  = {(const float*)d_in[4], (const float*)d_in[10],
                         (const float*)d_in[16], (const float*)d_in[22]};
  const float* Wd[3]  = {(const float*)d_in[5], (const float*)d_in[11], (const float*)d_in[17]};
  const float* bd[3]  = {(const float*)d_in[6], (const float*)d_in[12], (const float*)d_in[18]};
  const float* Ws[3]  = {(const float*)d_in[7], (const float*)d_in[13], (const float*)d_in[19]};
  const float* bs[3]  = {(const float*)d_in[8], (const float*)d_in[14], (const float*)d_in[20]};

  float* zbuf = (float*)d_ws;          // (B,N,C) pre-BN
  float* ybuf = zbuf + NTOT;           // (B,N,C) post-BN/leaky
  float* rbuf = ybuf + NTOT;           // (B,N) squared norms
  float* AB   = rbuf + NPIX;           // 32 scales + 32 shifts
  int*   idxb = (int*)(AB + 64);       // (B,N,8)

  conv_kernel<<<dim3(B_ * 32), dim3(256), 0, stream>>>(x, conv_w, conv_b, zbuf);
  stats_kernel<<<dim3(32), dim3(256), 0, stream>>>(zbuf, g[0], bn[0], AB);
  bnleaky_kernel<<<dim3(NPIX / 8), dim3(256), 0, stream>>>(zbuf, AB, ybuf, rbuf);
  knn_kernel<<<dim3(B_ * 64), dim3(32), 0, stream>>>(ybuf, rbuf, idxb);

  for (int l = 0; l < 3; ++l) {
    nla_kernel<<<dim3(B_ * 64), dim3(32), 0, stream>>>(ybuf, idxb, Wd[l], Ws[l],
                                                       bd[l], bs[l], zbuf);
    stats_kernel<<<dim3(32), dim3(256), 0, stream>>>(zbuf, g[l + 1], bn[l + 1], AB);
    if (l < 2)
      bnleaky_kernel<<<dim3(NPIX / 8), dim3(256), 0, stream>>>(zbuf, AB, ybuf, rbuf);
  }
  final_kernel<<<dim3(NTOT / 256), dim3(256), 0, stream>>>(x, zbuf, AB, (float*)d_out);
}